// ContrastiveLoss_69415261438629
// MI455X (gfx1250) — compile-verified
//
#include <hip/hip_runtime.h>
#include <stdint.h>

#define N_EMB 4096
#define D_EMB 768
#define KC 64
#define NCHUNK (D_EMB / KC)

typedef __attribute__((ext_vector_type(2))) float v2f;
typedef __attribute__((ext_vector_type(8))) float v8f;
typedef __attribute__((ext_vector_type(4))) unsigned int v4u;
typedef __attribute__((ext_vector_type(8))) unsigned int v8u;

// ---------------- Phase 1: per-row sum and sum-of-squares ----------------
__global__ void __launch_bounds__(256)
row_stats_kernel(const float* __restrict__ x,
                 float* __restrict__ sumsq,
                 float* __restrict__ sum1) {
  const int row = blockIdx.x;
  const int tid = threadIdx.x;
  const float* p = x + (size_t)row * D_EMB;
  float s2 = 0.f, s1 = 0.f;
  for (int i = tid; i < D_EMB; i += 256) {
    float v = p[i];
    s2 = fmaf(v, v, s2);
    s1 += v;
  }
  __shared__ float r2[256];
  __shared__ float r1[256];
  r2[tid] = s2;
  r1[tid] = s1;
  __syncthreads();
  for (int s = 128; s > 0; s >>= 1) {
    if (tid < s) { r2[tid] += r2[tid + s]; r1[tid] += r1[tid + s]; }
    __syncthreads();
  }
  if (tid == 0) { sumsq[row] = r2[0]; sum1[row] = r1[0]; }
}

// ---------------- TDM descriptor helpers (ISA 8.3 - 8.6) ----------------
// Group 0: [1:0]=count=1 | [63:32]=lds byte addr | [120:64]=global addr | [127:126]=type 2
__device__ __forceinline__ v4u tdm_group0(unsigned int lds_addr,
                                          unsigned long long gaddr) {
  v4u g;
  g.x = 1u;                                      // count=1, user descriptor
  g.y = lds_addr;                                // LDS byte address
  g.z = (unsigned int)(gaddr & 0xffffffffull);   // global addr [31:0]
  g.w = (unsigned int)((gaddr >> 32) & 0x01ffffffull)  // global addr [56:32]
      | (2u << 30);                              // type = 2 ("image")
  return g;
}

// Group 1 for a 2D fp32 tile: 32 rows x 64 dwords, row stride 768 elements,
// TDM pad feature recreates the [32][65] LDS layout (pad 1 dword / 64 dwords).
__device__ __forceinline__ v8u tdm_group1(void) {
  v8u g;
  g.s0 = (2u << 16)                    // data_size = 4 bytes
       | (1u << 20)                    // pad_enable
       | (5u << 22);                   // pad_interval: 64 dwords (2^(5+1))
                                       // pad_amount = 0 -> 1 dword
  g.s1 = ((unsigned)D_EMB & 0xffffu) << 16;     // tensor_dim0[15:0]  (bits 63:48)
  g.s2 = ((unsigned)D_EMB >> 16)                // tensor_dim0[31:16] (bits 79:64)
       | (((unsigned)N_EMB & 0xffffu) << 16);   // tensor_dim1[15:0]  (bits 95:80)
  g.s3 = ((unsigned)N_EMB >> 16)                // tensor_dim1[31:16] (bits 111:96)
       | ((unsigned)KC << 16);                  // tile_dim0 = 64     (bits 127:112)
  g.s4 = 32u;                                   // tile_dim1 = 32; tile_dim2 = 0
  g.s5 = (unsigned)D_EMB;                       // tensor_dim0_stride[31:0]
  g.s6 = 0u;                                    // stride[47:32]=0, dim1_stride=0
  g.s7 = 0u;
  return g;
}

// Issue one TDM tile load. Issued per-wave (EXEC ignored) -> call from one wave.
__device__ __forceinline__ void tdm_load(v4u g0, v8u g1) {
  v4u gz = {0u, 0u, 0u, 0u};  // zero-filled groups 2/3: tile_dim2/3 = 0 (2D)
  asm volatile("tensor_load_to_lds %0, %1, %2, %3"
               :
               : "s"(g0), "s"(g1), "s"(gz), "s"(gz)
               : "memory");
}

__device__ __forceinline__ unsigned int lds_byte_addr(const void* p) {
  // Generic pointers to LDS carry the LDS byte offset in the low 32 bits.
  return (unsigned int)(uintptr_t)p;
}

// ---------------- Phase 2: TDM-fed WMMA GEMM tile + fused loss ----------------
// 128 threads = 4 waves; 32x32 output tile, one 16x16 subtile per wave via
// V_WMMA_F32_16X16X4_F32. K staged through double-buffered LDS by the TDM.
__global__ void __launch_bounds__(128)
contrastive_tile_kernel(const float* __restrict__ img,
                        const float* __restrict__ txt,
                        const int* __restrict__ gt,
                        const float* __restrict__ sa,
                        const float* __restrict__ ra,
                        const float* __restrict__ sb,
                        const float* __restrict__ rb,
                        float* __restrict__ partials) {
  __shared__ float As[2][32][KC + 1];   // +1 pad written by TDM pad feature
  __shared__ float Bs[2][32][KC + 1];
  __shared__ float red[128];

  const int tid    = threadIdx.x;
  const int lane   = tid & 31;
  const int wave   = tid >> 5;
  const int tile_m = blockIdx.y * 32;
  const int tile_n = blockIdx.x * 32;

  const int m_off = (wave >> 1) * 16;
  const int n_off = (wave & 1) * 16;
  const int lhalf = lane >> 4;
  const int lm    = lane & 15;
  const int kb    = lhalf * 2;

  // Scalar (SGPR) leader test -> s_cbranch, so the TDM op issues exactly once.
  const bool leader = (__builtin_amdgcn_readfirstlane(tid) >> 5) == 0;

  const unsigned long long abase =
      (unsigned long long)(uintptr_t)(img + (size_t)tile_m * D_EMB);
  const unsigned long long bbase =
      (unsigned long long)(uintptr_t)(txt + (size_t)tile_n * D_EMB);
  const v8u g1 = tdm_group1();
  const unsigned int ldsA[2] = { lds_byte_addr(&As[0][0][0]),
                                 lds_byte_addr(&As[1][0][0]) };
  const unsigned int ldsB[2] = { lds_byte_addr(&Bs[0][0][0]),
                                 lds_byte_addr(&Bs[1][0][0]) };

  v8f acc = {0.f, 0.f, 0.f, 0.f, 0.f, 0.f, 0.f, 0.f};

  if (leader) {  // prologue: chunk 0 into buffer 0
    tdm_load(tdm_group0(ldsA[0], abase), g1);
    tdm_load(tdm_group0(ldsB[0], bbase), g1);
  }

  for (int chunk = 0; chunk < NCHUNK; ++chunk) {
    const int buf = chunk & 1;
    __builtin_amdgcn_s_wait_tensorcnt(0);  // wave0: chunk data landed in LDS
    __syncthreads();                       // publish to all waves

    if (leader && (chunk + 1 < NCHUNK)) {  // prefetch next chunk (overlaps WMMA)
      const unsigned long long koff = (unsigned long long)(chunk + 1) * KC * 4u;
      tdm_load(tdm_group0(ldsA[buf ^ 1], abase + koff), g1);
      tdm_load(tdm_group0(ldsB[buf ^ 1], bbase + koff), g1);
    }

    // 16 WMMAs per chunk. f32 fragment layout (ISA 7.12.2): lanes 0-15 hold
    // K=k+0,k+1 in (v0,v1); lanes 16-31 hold K=k+2,k+3.
#pragma unroll
    for (int k = 0; k < KC; k += 4) {
      v2f a, b;
      a.x = As[buf][m_off + lm][k + kb];
      a.y = As[buf][m_off + lm][k + kb + 1];
      b.x = Bs[buf][n_off + lm][k + kb];
      b.y = Bs[buf][n_off + lm][k + kb + 1];
      acc = __builtin_amdgcn_wmma_f32_16x16x4_f32(
          false, a, false, b, (short)0, acc, false, false);
    }
    __syncthreads();  // done reading buf before TDM overwrites it (chunk+2)
  }

  // Epilogue. C/D layout: VGPR r -> M = r + 8*lhalf, N = lane & 15.
  const int   nn    = tile_n + n_off + lm;
  const float sbn   = sb[nn];
  const float rbn   = rb[nn];
  const int   mrow0 = tile_m + m_off + lhalf * 8;

  float local = 0.f;
#pragma unroll
  for (int r = 0; r < 8; ++r) {
    const int   mm   = mrow0 + r;
    const float dotv = acc[r];
    float sq = sa[mm] + sbn - 2.0f * dotv
             + 2.0e-6f * (ra[mm] - rbn)
             + (float)D_EMB * 1.0e-12f;
    sq = fmaxf(sq, 0.0f);               // dist^2 = max(sq, 0)
    const float dist = sqrtf(sq);
    const float lab  = (float)gt[(size_t)mm * N_EMB + nn];
    const float h    = fmaxf(2.0f - dist, 0.0f);
    local += (1.0f - lab) * sq + lab * h * h;
  }

  red[tid] = local;
  __syncthreads();
  for (int s = 64; s > 0; s >>= 1) {
    if (tid < s) red[tid] += red[tid + s];
    __syncthreads();
  }
  if (tid == 0) partials[blockIdx.y * gridDim.x + blockIdx.x] = red[0];
}

// ---------------- Phase 3: final reduction (double precision) ----------------
__global__ void __launch_bounds__(256)
finalize_kernel(const float* __restrict__ partials, int num,
                float* __restrict__ out) {
  const int tid = threadIdx.x;
  double acc = 0.0;
  for (int i = tid; i < num; i += 256) acc += (double)partials[i];
  __shared__ double red[256];
  red[tid] = acc;
  __syncthreads();
  for (int s = 128; s > 0; s >>= 1) {
    if (tid < s) red[tid] += red[tid + s];
    __syncthreads();
  }
  if (tid == 0)
    out[0] = (float)(red[0] * (1.0 / ((double)N_EMB * (double)N_EMB)));
}

// ---------------- Host launcher ----------------
extern "C" void kernel_launch(void* const* d_in, const int* in_sizes, int n_in,
                              void* d_out, int out_size, void* d_ws, size_t ws_size,
                              hipStream_t stream) {
  (void)in_sizes; (void)n_in; (void)out_size; (void)ws_size;
  const float* img = (const float*)d_in[0];
  const float* txt = (const float*)d_in[1];
  const int*   gt  = (const int*)d_in[2];

  float* ws = (float*)d_ws;
  float* sa = ws;                 // 4096
  float* ra = ws + 4096;          // 4096
  float* sb = ws + 8192;          // 4096
  float* rb = ws + 12288;         // 4096
  float* partials = ws + 16384;   // 128*128 = 16384 floats

  row_stats_kernel<<<N_EMB, 256, 0, stream>>>(img, sa, ra);
  row_stats_kernel<<<N_EMB, 256, 0, stream>>>(txt, sb, rb);

  dim3 grid(N_EMB / 32, N_EMB / 32);  // 128 x 128 tiles
  contrastive_tile_kernel<<<grid, 128, 0, stream>>>(img, txt, gt,
                                                    sa, ra, sb, rb, partials);

  finalize_kernel<<<1, 256, 0, stream>>>(partials, (N_EMB / 32) * (N_EMB / 32),
                                         (float*)d_out);
}